// MorphModule_80118319940266
// MI455X (gfx1250) — compile-verified
//
#include <hip/hip_runtime.h>
#include <stdint.h>

// Grayscale 5x5 erosion, border fill = 1e20.
// VALU-bound (~38 f32 ops/px > HBM floor), so: min3-friendly balanced
// reduction, sliding-row register buffer (16 ds_load_b128/thread), and
// CDNA5 async global->LDS b64 staging (ASYNCcnt) for the haloed tile.

#define BORDER 1e20f

constexpr int BATCH  = 16;
constexpr int H      = 1024;
constexpr int W      = 1024;
constexpr int TILE_W = 256;          // outputs per block, x
constexpr int TILE_H = 16;           // outputs per block, y
constexpr int HALO   = 2;            // 5x5 footprint, origin (2,2)
constexpr int IN_W   = TILE_W + 2 * HALO;   // 260 input cols incl. halo
constexpr int IN_H   = TILE_H + 2 * HALO;   // 20 input rows incl. halo
constexpr int IN_W2  = IN_W / 2;            // 130 column pairs per row
constexpr int LW     = 264;          // LDS row stride in floats (mult of 8)
constexpr int LW4    = LW / 4;       // 66 float4 per LDS row
constexpr int NTHREADS = 256;        // 8 wave32

__global__ __launch_bounds__(NTHREADS)
void MorphModule_erode5x5_kernel(const float* __restrict__ img,
                                 const float* __restrict__ se,
                                 float* __restrict__ out)
{
    __shared__ float tile[IN_H * LW];   // 20*264*4 = 21120 B LDS

    const int tid = threadIdx.x;
    const int gx0 = blockIdx.x * TILE_W;       // tile origin x
    const int gy0 = blockIdx.y * TILE_H;       // tile origin y
    const int bb  = blockIdx.z;                // batch index
    const float* __restrict__ imgB = img + (size_t)bb * H * W;

    // ------------------------------------------------------------------
    // Stage haloed input tile into LDS as 8-byte column pairs.
    // Halo offset is -2, W is even -> global col (gx0-2+2p) is even, so a
    // pair is 8B-aligned in global AND in LDS, and is always entirely
    // in-bounds or entirely out-of-bounds in x.
    // In-bounds pairs: CDNA5 async copy engine (ASYNCcnt, EXEC-predicated).
    // OOB pairs: plain 8B DS store of {BORDER,BORDER} (disjoint addresses).
    // ------------------------------------------------------------------
    for (int idx = tid; idx < IN_H * IN_W2; idx += NTHREADS) {
        const int ly = idx / IN_W2;
        const int p  = idx - ly * IN_W2;
        const int lx = 2 * p;
        const int gy = gy0 - HALO + ly;
        const int gx = gx0 - HALO + lx;
        float* cell = &tile[ly * LW + lx];
        if ((unsigned)gy < (unsigned)H && (unsigned)gx < (unsigned)W) {
            const float* g = imgB + (size_t)gy * W + gx;
            uint32_t lds_off = (uint32_t)(uintptr_t)cell;  // low 32 = LDS byte off
            asm volatile("global_load_async_to_lds_b64 %0, %1, off"
                         :: "v"(lds_off), "v"(g)
                         : "memory");
        } else {
            *(float2*)cell = make_float2(BORDER, BORDER);
        }
    }

    asm volatile("s_wait_asynccnt 0x0" ::: "memory");
    __syncthreads();

    // 5x5 structuring element: wave-uniform address -> scalar loads/SGPRs.
    float s[25];
#pragma unroll
    for (int t = 0; t < 25; ++t) s[t] = se[t];

    // ------------------------------------------------------------------
    // Compute: thread (tx, ty) produces float4 column tx for 4 consecutive
    // output rows r0..r0+3.  Sliding 5-row register window: preload rows
    // r0..r0+4, stream one new row per output row -> 16 ds_load_b128 total.
    // ------------------------------------------------------------------
    const int tx = tid & 63;    // float4 column 0..63
    const int ty = tid >> 6;    // 0..3
    const int r0 = ty * 4;      // first output row of this thread
    const float4* __restrict__ t4 = (const float4*)tile;

    float4 blo[5], bhi[5];
#pragma unroll
    for (int m = 0; m < 5; ++m) {
        blo[m] = t4[(r0 + m) * LW4 + tx];
        bhi[m] = t4[(r0 + m) * LW4 + tx + 1];
    }

#pragma unroll
    for (int k = 0; k < 4; ++k) {
        // Window rows for output row r0+k are image rows r0+k .. r0+k+4,
        // living in buffer slots (k+i) % 5 (constants after unroll).
        float w[5][8];
#pragma unroll
        for (int i = 0; i < 5; ++i) {
            const int slot = (k + i) % 5;
            const float4 lo = blo[slot], hi = bhi[slot];
            w[i][0] = lo.x; w[i][1] = lo.y; w[i][2] = lo.z; w[i][3] = lo.w;
            w[i][4] = hi.x; w[i][5] = hi.y; w[i][6] = hi.z; w[i][7] = hi.w;
        }

        float res[4];
#pragma unroll
        for (int c = 0; c < 4; ++c) {
            float t[25];
#pragma unroll
            for (int i = 0; i < 5; ++i)
#pragma unroll
                for (int j = 0; j < 5; ++j)
                    t[i * 5 + j] = w[i][c + j] - s[i * 5 + j];

            // Balanced groups of 3 -> backend folds into v_min3_num_f32.
            const float g0 = fminf(fminf(t[0],  t[1]),  t[2]);
            const float g1 = fminf(fminf(t[3],  t[4]),  t[5]);
            const float g2 = fminf(fminf(t[6],  t[7]),  t[8]);
            const float g3 = fminf(fminf(t[9],  t[10]), t[11]);
            const float g4 = fminf(fminf(t[12], t[13]), t[14]);
            const float g5 = fminf(fminf(t[15], t[16]), t[17]);
            const float g6 = fminf(fminf(t[18], t[19]), t[20]);
            const float g7 = fminf(fminf(t[21], t[22]), t[23]);
            const float h0 = fminf(fminf(g0, g1), g2);
            const float h1 = fminf(fminf(g3, g4), g5);
            const float h2 = fminf(fminf(g6, g7), t[24]);
            // Final clamp reproduces reference's BORDER-initialized min.
            res[c] = fminf(fminf(fminf(h0, h1), h2), BORDER);
        }

        float4* o = (float4*)(out + ((size_t)bb * H + gy0 + r0 + k) * W
                              + gx0 + tx * 4);
        *o = make_float4(res[0], res[1], res[2], res[3]);  // b128, coalesced

        if (k < 3) {  // stream in image row r0+k+5 -> slot (k+5)%5 == k
            blo[k] = t4[(r0 + k + 5) * LW4 + tx];
            bhi[k] = t4[(r0 + k + 5) * LW4 + tx + 1];
        }
    }
}

extern "C" void kernel_launch(void* const* d_in, const int* in_sizes, int n_in,
                              void* d_out, int out_size, void* d_ws, size_t ws_size,
                              hipStream_t stream) {
    const float* img = (const float*)d_in[0];   // (16, 1024, 1024) f32
    const float* se  = (const float*)d_in[1];   // (5, 5) f32
    float* out = (float*)d_out;                 // (16, 1024, 1024) f32

    dim3 grid(W / TILE_W, H / TILE_H, BATCH);   // (4, 64, 16)
    MorphModule_erode5x5_kernel<<<grid, NTHREADS, 0, stream>>>(img, se, out);
}